// RoiPoolingConv_55757265437286
// MI455X (gfx1250) — compile-verified
//
#include <hip/hip_runtime.h>
#include <hip/hip_bf16.h>

#define POOL 7
#define CELLS (POOL * POOL * POOL)   // 343
#define CH 128
#define GRIDL 96
#define PLANE (GRIDL * GRIDL)        // 9216
#define VOL ((size_t)GRIDL * GRIDL * GRIDL)  // 884736
#define CH_PER_BLOCK 16
#define NELEM (CH_PER_BLOCK * CELLS) // 5488
#define BLOCK 256                    // 8 wave32s

#if defined(__gfx1250__) && __has_builtin(__builtin_amdgcn_global_load_async_to_lds_b32)
#define USE_ASYNC_LDS 1
#else
#define USE_ASYNC_LDS 0
#endif

__global__ __launch_bounds__(BLOCK) void roi_pool3d_gfx1250(
    const float* __restrict__ feat,   // (C, X, Y, Z) fp32
    const int*   __restrict__ rois,   // (R, 6) = x,y,z,w,h,d
    float*       __restrict__ out)    // (R, C, 7,7,7)
{
    __shared__ int   s_c0[3][POOL];
    __shared__ int   s_c1[3][POOL];
    __shared__ float s_f [3][POOL];
    __shared__ int   s_roi[8];        // ROI descriptor staged via async DMA

    const int chunks = CH / CH_PER_BLOCK;
    const int r      = blockIdx.x / chunks;
    const int cbase  = (blockIdx.x % chunks) * CH_PER_BLOCK;
    const int tid    = threadIdx.x;

    // ---- stage the ROI descriptor (6 ints) into LDS via gfx1250 async DMA ----
    // Issuing lanes (tid<6) and consuming lanes (tid<21) are all in wave 0, so
    // s_wait_asynccnt==0 is sufficient ordering (no extra barrier needed before
    // the table computation; other waves sync at __syncthreads below).
#if USE_ASYNC_LDS
    if (tid < 6) {
        int* gsrc = (int*)(rois + r * 6 + tid);   // const stripped for builtin
        __builtin_amdgcn_global_load_async_to_lds_b32(
            (__attribute__((address_space(1))) int*)gsrc,
            (__attribute__((address_space(3))) int*)(&s_roi[tid]),
            0, 0);
    }
#if __has_builtin(__builtin_amdgcn_s_wait_asynccnt)
    __builtin_amdgcn_s_wait_asynccnt(0);
#endif
#define ROI_VAL(k) ((float)s_roi[(k)])
#else
#define ROI_VAL(k) ((float)rois[r * 6 + (k)])
#endif

    // ---- per-ROI interpolation tables (21 threads: axis in {x,y,z}, i in 0..6) ----
    if (tid < 3 * POOL) {
        const int axis = tid / POOL;
        const int i    = tid % POOL;
        const float start = ROI_VAL(axis);
        const float size  = ROI_VAL(3 + axis);
        // reference: coords = start + (i * size) / pool_size
        const float coord = start + ((float)i * size) / (float)POOL;
        int c0 = (int)floorf(coord);
        if (c0 < 0) c0 = 0;
        if (c0 > GRIDL - 1) c0 = GRIDL - 1;
        int c1 = c0 + 1;
        if (c1 > GRIDL - 1) c1 = GRIDL - 1;
        s_c0[axis][i] = c0;
        s_c1[axis][i] = c1;
        s_f [axis][i] = coord - (float)c0;   // frac against clipped floor, like reference
    }
    __syncthreads();

    // ---- main gather/blend loop: 16 channels x 343 cells per block ----
    for (int e = tid; e < NELEM; e += BLOCK) {
        const int c    = cbase + e / CELLS;
        const int cell = e % CELLS;
        const int px   = cell / (POOL * POOL);
        const int py   = (cell / POOL) % POOL;
        const int pz   = cell % POOL;

        const int x0 = s_c0[0][px], x1 = s_c1[0][px];
        const int y0 = s_c0[1][py], y1 = s_c1[1][py];
        const int z0 = s_c0[2][pz], z1 = s_c1[2][pz];
        const float fx = s_f[0][px], fy = s_f[1][py], fz = s_f[2][pz];
        const float gx = 1.0f - fx, gy = 1.0f - fy, gz = 1.0f - fz;

        const float* __restrict__ base = feat + (size_t)c * VOL;
        const int a0 = x0 * PLANE, a1 = x1 * PLANE;
        const int b0 = y0 * GRIDL, b1 = y1 * GRIDL;

        // 8 independent gathers -> 8 outstanding global_load_b32 per lane
        const float v000 = base[a0 + b0 + z0];
        const float v100 = base[a1 + b0 + z0];
        const float v010 = base[a0 + b1 + z0];
        const float v110 = base[a1 + b1 + z0];
        const float v001 = base[a0 + b0 + z1];
        const float v101 = base[a1 + b0 + z1];
        const float v011 = base[a0 + b1 + z1];
        const float v111 = base[a1 + b1 + z1];

        // prefetch the same corner one channel ahead (speculative)
        if (c + 1 < CH) {
            __builtin_prefetch(base + VOL + (size_t)(a0 + b0 + z0), 0, 1);
        }

        // product-form blend to mirror the reference's rounding behavior
        const float gygz = gy * gz, fygz = fy * gz, gyfz = gy * fz, fyfz = fy * fz;
        const float w =
              v000 * (gx * gygz) + v100 * (fx * gygz)
            + v010 * (gx * fygz) + v110 * (fx * fygz)
            + v001 * (gx * gyfz) + v101 * (fx * gyfz)
            + v011 * (gx * fyfz) + v111 * (fx * fyfz);

        out[((size_t)r * CH + c) * CELLS + cell] = w;
    }
}

extern "C" void kernel_launch(void* const* d_in, const int* in_sizes, int n_in,
                              void* d_out, int out_size, void* d_ws, size_t ws_size,
                              hipStream_t stream) {
    (void)n_in; (void)out_size; (void)d_ws; (void)ws_size;
    const float* img  = (const float*)d_in[0];
    const int*   rois = (const int*)d_in[1];
    float*       out  = (float*)d_out;

    const int R = in_sizes[1] / 6;                 // 128 ROIs
    dim3 grid(R * (CH / CH_PER_BLOCK));            // 128 * 8 = 1024 blocks
    dim3 block(BLOCK);                              // 256 threads = 8 wave32s
    roi_pool3d_gfx1250<<<grid, block, 0, stream>>>(img, rois, out);
}